// DGCN_AE_79860621902496
// MI455X (gfx1250) — compile-verified
//
#include <hip/hip_runtime.h>
#include <hip/hip_bf16.h>

// ---------------------------------------------------------------------------
// MI455X (gfx1250) implementation of the DGCN-AE forward pass.
// All heavy GEMMs use V_WMMA_F32_16X16X4_F32 (exact fp32 matrix core path).
// Everything is L2-resident (~130MB). Inner loops are branch-free (boundary
// handling via index clamping: garbage rows/cols never cross-contaminate a
// WMMA tile and are discarded by guarded stores). Wave tile is 32x64
// (8 accumulators) so each B fragment is reused by two A fragments.
// ---------------------------------------------------------------------------

typedef float v2f __attribute__((ext_vector_type(2)));
typedef float v8f __attribute__((ext_vector_type(8)));

#define NNODES 1024
#define BATCH  8
#define CIN    288
#define ZDIM   128
#define NM     5
#define LSTMH  144
#define GATES  576

__device__ __forceinline__ v8f wmma_f32(v2f a, v2f b, v8f c) {
  // D = A(16x4) * B(4x16) + C(16x16), fp32, wave32
  return __builtin_amdgcn_wmma_f32_16x16x4_f32(false, a, false, b, (short)0, c,
                                               false, false);
}

__device__ __forceinline__ float sigm(float x) { return 1.0f / (1.0f + __expf(-x)); }

// ---------------------------------------------------------------------------
// Generic batched strided GEMM:  C[z] = alpha * A[z] @ B[z] + beta * Cin[z]
// A element (m,k) at A + z*bsA + m*rsA + k*csA   (transpose == swapped strides)
// B element (k,n) at B + z*bsB + k*rsB + n*csB
// Block: 128 threads = 4 waves; block tile 128(M) x 64(N); wave tile 32x64.
// Out-of-range rows/cols are CLAMPED (loads always valid, no EXEC divergence);
// their accumulator lanes are discarded by the guarded store.
// ---------------------------------------------------------------------------
__global__ void __launch_bounds__(128)
gemm_wmma_kernel(const float* __restrict__ A, long rsA, long csA, long bsA,
                 const float* __restrict__ B, long rsB, long csB, long bsB,
                 float* __restrict__ C, long ldc, long bsC,
                 const float* __restrict__ Cin, long ldcin, long bsCin,
                 float alpha, float beta, int M, int N, int K)
{
  const int lane = threadIdx.x & 31;
  const int wave = threadIdx.x >> 5;
  const int half = lane >> 4;        // selects K pair {2h, 2h+1}
  const int lm   = lane & 15;

  const long z = blockIdx.z;
  A += z * bsA;
  B += z * bsB;
  C += z * bsC;
  const float* Ci = Cin ? (Cin + z * bsCin) : nullptr;

  const int m0 = blockIdx.y * 128 + wave * 32;   // 32 rows per wave
  const int n0 = blockIdx.x * 64;

  const int mLo = min(m0 + lm, M - 1);           // clamped A rows
  const int mHi = min(m0 + 16 + lm, M - 1);
  const float* ArowLo = A + (long)mLo * rsA;
  const float* ArowHi = A + (long)mHi * rsA;

  int ncol[4];
  const float* Bcol[4];
#pragma unroll
  for (int t = 0; t < 4; ++t) {
    ncol[t] = n0 + t * 16 + lm;
    Bcol[t] = B + (long)min(ncol[t], N - 1) * csB;   // clamped B col
  }

  v8f accLo[4], accHi[4];
#pragma unroll
  for (int t = 0; t < 4; ++t) {
    accLo[t] = (v8f){0.f, 0.f, 0.f, 0.f, 0.f, 0.f, 0.f, 0.f};
    accHi[t] = (v8f){0.f, 0.f, 0.f, 0.f, 0.f, 0.f, 0.f, 0.f};
  }

  const int ka0 = 2 * half;

  if (csA == 1 && rsB == 1) {
    // Both operands contiguous along K: pure v2f (b64) loads, 1-deep pipeline.
    const float* apL = ArowLo + ka0;
    const float* apH = ArowHi + ka0;
    const float* bp0 = Bcol[0] + ka0;
    const float* bp1 = Bcol[1] + ka0;
    const float* bp2 = Bcol[2] + ka0;
    const float* bp3 = Bcol[3] + ka0;
    v2f aL = *(const v2f*)apL, aH = *(const v2f*)apH;
    v2f b0 = *(const v2f*)bp0, b1 = *(const v2f*)bp1;
    v2f b2 = *(const v2f*)bp2, b3 = *(const v2f*)bp3;
    for (int k = K - 4; k > 0; k -= 4) {
      apL += 4; apH += 4; bp0 += 4; bp1 += 4; bp2 += 4; bp3 += 4;
      v2f anL = *(const v2f*)apL, anH = *(const v2f*)apH;
      v2f bn0 = *(const v2f*)bp0, bn1 = *(const v2f*)bp1;
      v2f bn2 = *(const v2f*)bp2, bn3 = *(const v2f*)bp3;
      accLo[0] = wmma_f32(aL, b0, accLo[0]);
      accHi[0] = wmma_f32(aH, b0, accHi[0]);
      accLo[1] = wmma_f32(aL, b1, accLo[1]);
      accHi[1] = wmma_f32(aH, b1, accHi[1]);
      accLo[2] = wmma_f32(aL, b2, accLo[2]);
      accHi[2] = wmma_f32(aH, b2, accHi[2]);
      accLo[3] = wmma_f32(aL, b3, accLo[3]);
      accHi[3] = wmma_f32(aH, b3, accHi[3]);
      aL = anL; aH = anH; b0 = bn0; b1 = bn1; b2 = bn2; b3 = bn3;
    }
    accLo[0] = wmma_f32(aL, b0, accLo[0]);
    accHi[0] = wmma_f32(aH, b0, accHi[0]);
    accLo[1] = wmma_f32(aL, b1, accLo[1]);
    accHi[1] = wmma_f32(aH, b1, accHi[1]);
    accLo[2] = wmma_f32(aL, b2, accLo[2]);
    accHi[2] = wmma_f32(aH, b2, accHi[2]);
    accLo[3] = wmma_f32(aL, b3, accLo[3]);
    accHi[3] = wmma_f32(aH, b3, accHi[3]);
  } else if (csA == 1) {
    // A contiguous (v2f); B strided along K (row-major B): scalar pair loads.
    const float* apL = ArowLo + ka0;
    const float* apH = ArowHi + ka0;
    const float* bp0 = Bcol[0] + (long)ka0 * rsB;
    const float* bp1 = Bcol[1] + (long)ka0 * rsB;
    const float* bp2 = Bcol[2] + (long)ka0 * rsB;
    const float* bp3 = Bcol[3] + (long)ka0 * rsB;
    const long step = 4 * rsB;
    v2f aL = *(const v2f*)apL, aH = *(const v2f*)apH;
    v2f b0, b1, b2, b3;
    b0.x = bp0[0]; b0.y = bp0[rsB];
    b1.x = bp1[0]; b1.y = bp1[rsB];
    b2.x = bp2[0]; b2.y = bp2[rsB];
    b3.x = bp3[0]; b3.y = bp3[rsB];
    for (int k = K - 4; k > 0; k -= 4) {
      apL += 4; apH += 4; bp0 += step; bp1 += step; bp2 += step; bp3 += step;
      v2f anL = *(const v2f*)apL, anH = *(const v2f*)apH;
      v2f bn0, bn1, bn2, bn3;
      bn0.x = bp0[0]; bn0.y = bp0[rsB];
      bn1.x = bp1[0]; bn1.y = bp1[rsB];
      bn2.x = bp2[0]; bn2.y = bp2[rsB];
      bn3.x = bp3[0]; bn3.y = bp3[rsB];
      accLo[0] = wmma_f32(aL, b0, accLo[0]);
      accHi[0] = wmma_f32(aH, b0, accHi[0]);
      accLo[1] = wmma_f32(aL, b1, accLo[1]);
      accHi[1] = wmma_f32(aH, b1, accHi[1]);
      accLo[2] = wmma_f32(aL, b2, accLo[2]);
      accHi[2] = wmma_f32(aH, b2, accHi[2]);
      accLo[3] = wmma_f32(aL, b3, accLo[3]);
      accHi[3] = wmma_f32(aH, b3, accHi[3]);
      aL = anL; aH = anH; b0 = bn0; b1 = bn1; b2 = bn2; b3 = bn3;
    }
    accLo[0] = wmma_f32(aL, b0, accLo[0]);
    accHi[0] = wmma_f32(aH, b0, accHi[0]);
    accLo[1] = wmma_f32(aL, b1, accLo[1]);
    accHi[1] = wmma_f32(aH, b1, accHi[1]);
    accLo[2] = wmma_f32(aL, b2, accLo[2]);
    accHi[2] = wmma_f32(aH, b2, accHi[2]);
    accLo[3] = wmma_f32(aL, b3, accLo[3]);
    accHi[3] = wmma_f32(aH, b3, accHi[3]);
  } else {
    // Fully generic fallback (unused by current call sites).
    for (int k = 0; k < K; k += 4) {
      const int kk = k + ka0;
      v2f aL, aH;
      aL.x = ArowLo[(long)kk * csA];
      aL.y = ArowLo[(long)(kk + 1) * csA];
      aH.x = ArowHi[(long)kk * csA];
      aH.y = ArowHi[(long)(kk + 1) * csA];
#pragma unroll
      for (int t = 0; t < 4; ++t) {
        v2f b;
        b.x = Bcol[t][(long)kk * rsB];
        b.y = Bcol[t][(long)(kk + 1) * rsB];
        accLo[t] = wmma_f32(aL, b, accLo[t]);
        accHi[t] = wmma_f32(aH, b, accHi[t]);
      }
    }
  }

#pragma unroll
  for (int t = 0; t < 4; ++t) {
    const int n = ncol[t];
    if (n >= N) continue;
#pragma unroll
    for (int v = 0; v < 8; ++v) {
      const int m = m0 + v + half * 8;   // C/D: VGPR v -> rows v / v+8
      if (m < M) {
        float val = alpha * accLo[t][v];
        if (Ci) val += beta * Ci[(long)m * ldcin + n];
        C[(long)m * ldc + n] = val;
      }
      const int mh = m + 16;
      if (mh < M) {
        float val = alpha * accHi[t][v];
        if (Ci) val += beta * Ci[(long)mh * ldcin + n];
        C[(long)mh * ldc + n] = val;
      }
    }
  }
}

// ---------------------------------------------------------------------------
// Elementwise epilogue: C = act(C + bias[col]) + res ; act: 0=none 1=relu
// 2=sigmoid 3=elu
// ---------------------------------------------------------------------------
__global__ void epilogue_kernel(float* __restrict__ C, const float* __restrict__ bias,
                                const float* __restrict__ res, long n, int cols, int act)
{
  long idx = (long)blockIdx.x * blockDim.x + threadIdx.x;
  if (idx >= n) return;
  float v = C[idx];
  if (bias) v += bias[idx % cols];
  if (act == 1)      v = fmaxf(v, 0.0f);
  else if (act == 2) v = 1.0f / (1.0f + __expf(-v));
  else if (act == 3) v = (v > 0.0f) ? v : (__expf(v) - 1.0f);
  if (res) v += res[idx];
  C[idx] = v;
}

// ---------------------------------------------------------------------------
// Conv1d (CIN=288 -> 64, k=3, pad=1) + ReLU, output transposed to [B,N,64]
// ---------------------------------------------------------------------------
__global__ void conv_kernel(const float* __restrict__ X, const float* __restrict__ W,
                            const float* __restrict__ bias, float* __restrict__ out)
{
  long idx = (long)blockIdx.x * blockDim.x + threadIdx.x;
  if (idx >= (long)BATCH * NNODES * 64) return;
  const int oc = idx % 64;
  const int n  = (idx / 64) % NNODES;
  const int b  = idx / (64 * NNODES);
  const float* xb = X + (long)b * CIN * NNODES;
  const float* wo = W + (long)oc * CIN * 3;
  float s = bias[oc];
  for (int ic = 0; ic < CIN; ++ic) {
    const float* xr = xb + (long)ic * NNODES;
    const float* wr = wo + ic * 3;
    if (n > 0)          s += xr[n - 1] * wr[0];
    s += xr[n] * wr[1];
    if (n < NNODES - 1) s += xr[n + 1] * wr[2];
  }
  out[((long)b * NNODES + n) * 64 + oc] = fmaxf(s, 0.0f);
}

// ---------------------------------------------------------------------------
// One BiLSTM layer. grid = 2 blocks (dir 0 fwd / 1 bwd), 288 threads (9 waves).
// h/c/gates resident in LDS; per-step gate GEMM via fp32 WMMA
// (M=8 clamped to 16, N=576 = 9 waves x 64, K = I then 144).
// Rows 8..15 are clamped duplicates of row 7 -> never stored.
// Output written into out[B,T,288] at feature offset dir*144.
// ---------------------------------------------------------------------------
__global__ void __launch_bounds__(288)
lstm_layer_kernel(const float* __restrict__ x, int I,
                  const float* __restrict__ Wih0, const float* __restrict__ Whh0,
                  const float* __restrict__ bih0, const float* __restrict__ bhh0,
                  const float* __restrict__ Wih1, const float* __restrict__ Whh1,
                  const float* __restrict__ bih1, const float* __restrict__ bhh1,
                  float* __restrict__ out)
{
  const int dir = blockIdx.x;
  const float* Wih = dir ? Wih1 : Wih0;
  const float* Whh = dir ? Whh1 : Whh0;
  const float* bih = dir ? bih1 : bih0;
  const float* bhh = dir ? bhh1 : bhh0;

  __shared__ float sh[BATCH * LSTMH];
  __shared__ float sc[BATCH * LSTMH];
  __shared__ float sg[BATCH * GATES];

  for (int u = threadIdx.x; u < BATCH * LSTMH; u += 288) { sh[u] = 0.f; sc[u] = 0.f; }
  __syncthreads();

  const int lane = threadIdx.x & 31;
  const int wave = threadIdx.x >> 5;
  const int half = lane >> 4;
  const int lm   = lane & 15;
  const int mrow = min(lm, 7);              // clamp batch row (8..15 discarded)
  const int ka0  = 2 * half;

  int ntile[4];
#pragma unroll
  for (int q = 0; q < 4; ++q) ntile[q] = wave * 64 + q * 16 + lm;   // < 576 always

  for (int s = 0; s < NNODES; ++s) {
    const int t = dir ? (NNODES - 1 - s) : s;

    v8f acc[4];
#pragma unroll
    for (int q = 0; q < 4; ++q)
      acc[q] = (v8f){0.f, 0.f, 0.f, 0.f, 0.f, 0.f, 0.f, 0.f};

    // ---- K over input features (x_t @ Wih^T) ----
    const float* xr = x + ((long)mrow * NNODES + t) * I + ka0;
#pragma unroll 2
    for (int k = 0; k < I; k += 4) {
      v2f a = *(const v2f*)(xr + k);
#pragma unroll
      for (int q = 0; q < 4; ++q) {
        v2f b = *(const v2f*)(&Wih[(long)ntile[q] * I + k + ka0]);
        acc[q] = wmma_f32(a, b, acc[q]);
      }
    }
    // ---- K over hidden state (h @ Whh^T), A-frag from LDS ----
#pragma unroll 2
    for (int k = 0; k < LSTMH; k += 4) {
      v2f a = *(const v2f*)(&sh[mrow * LSTMH + k + ka0]);
#pragma unroll
      for (int q = 0; q < 4; ++q) {
        v2f b = *(const v2f*)(&Whh[(long)ntile[q] * LSTMH + k + ka0]);
        acc[q] = wmma_f32(a, b, acc[q]);
      }
    }
    // write gates to LDS (rows 0..7 only)
#pragma unroll
    for (int q = 0; q < 4; ++q) {
#pragma unroll
      for (int v = 0; v < 8; ++v) {
        const int m = v + half * 8;
        if (m < 8) sg[m * GATES + ntile[q]] = acc[q][v];
      }
    }
    __syncthreads();

    // pointwise cell update (torch gate order i,f,g,o)
    for (int u = threadIdx.x; u < BATCH * LSTMH; u += 288) {
      const int b = u / LSTMH, j = u % LSTMH;
      const float gi = sg[b * GATES + j]             + bih[j]             + bhh[j];
      const float gf = sg[b * GATES + LSTMH + j]     + bih[LSTMH + j]     + bhh[LSTMH + j];
      const float gg = sg[b * GATES + 2 * LSTMH + j] + bih[2 * LSTMH + j] + bhh[2 * LSTMH + j];
      const float go = sg[b * GATES + 3 * LSTMH + j] + bih[3 * LSTMH + j] + bhh[3 * LSTMH + j];
      const float cn = sigm(gf) * sc[u] + sigm(gi) * tanhf(gg);
      sc[u] = cn;
      const float hn = sigm(go) * tanhf(cn);
      sh[u] = hn;
      out[((long)b * NNODES + t) * 288 + dir * LSTMH + j] = hn;
    }
    __syncthreads();
  }
}

// ---------------------------------------------------------------------------
// GAT attention scores: per (b,i) compute s = h @ a_s, n = h @ a_n
// ---------------------------------------------------------------------------
__global__ void sva_kernel(const float* __restrict__ h, const float* __restrict__ a_s,
                           const float* __restrict__ a_n, float* __restrict__ ss,
                           float* __restrict__ sn, int F)
{
  long idx = (long)blockIdx.x * blockDim.x + threadIdx.x;
  if (idx >= (long)BATCH * NNODES) return;
  const float* hp = h + idx * F;
  float s1 = 0.f, s2 = 0.f;
  for (int f = 0; f < F; ++f) { s1 += hp[f] * a_s[f]; s2 += hp[f] * a_n[f]; }
  ss[idx] = s1;
  sn[idx] = s2;
}

// ---------------------------------------------------------------------------
// GAT row: e = leaky_relu((ss[i]+sn[j]) * M[i,j]); mask by adj; row softmax.
// grid (1024 rows, 8 batch), 256 threads.
// ---------------------------------------------------------------------------
__global__ void __launch_bounds__(256)
gat_att_kernel(const float* __restrict__ ss, const float* __restrict__ sn,
               const float* __restrict__ Mm, const float* __restrict__ adj,
               float* __restrict__ att)
{
  const int i = blockIdx.x, b = blockIdx.y, tid = threadIdx.x;
  const float si = ss[b * NNODES + i];
  float* arow = att + ((long)b * NNODES + i) * NNODES;
  __shared__ float red[256];

  float lmax = -3.0e38f;
  for (int j = tid; j < NNODES; j += 256) {
    float e = (si + sn[b * NNODES + j]) * Mm[(long)i * NNODES + j];
    e = (e > 0.f) ? e : 0.2f * e;
    const float val = (adj[(long)i * NNODES + j] > 0.f) ? e : -9.0e15f;
    arow[j] = val;
    lmax = fmaxf(lmax, val);
  }
  red[tid] = lmax; __syncthreads();
  for (int st = 128; st > 0; st >>= 1) {
    if (tid < st) red[tid] = fmaxf(red[tid], red[tid + st]);
    __syncthreads();
  }
  const float mx = red[0]; __syncthreads();

  float lsum = 0.f;
  for (int j = tid; j < NNODES; j += 256) {
    const float ex = __expf(arow[j] - mx);
    arow[j] = ex;
    lsum += ex;
  }
  red[tid] = lsum; __syncthreads();
  for (int st = 128; st > 0; st >>= 1) {
    if (tid < st) red[tid] += red[tid + st];
    __syncthreads();
  }
  const float inv = 1.0f / red[0];
  for (int j = tid; j < NNODES; j += 256) arow[j] *= inv;
}

// ---------------------------------------------------------------------------
// Row L2-normalize [rows x 128] (cols == blockDim.x)
// ---------------------------------------------------------------------------
__global__ void __launch_bounds__(128)
rownorm_kernel(const float* __restrict__ in, float* __restrict__ out, int cols)
{
  const long row = blockIdx.x;
  const int tid = threadIdx.x;
  __shared__ float red[128];
  const float v = in[row * cols + tid];
  red[tid] = v * v; __syncthreads();
  for (int st = 64; st > 0; st >>= 1) {
    if (tid < st) red[tid] += red[tid + st];
    __syncthreads();
  }
  const float inv = 1.0f / fmaxf(sqrtf(red[0]), 1e-12f);
  out[row * cols + tid] = v * inv;
}

// ---------------------------------------------------------------------------
// MHA core: per (l,n,head) thread; L=8, heads=8, hd=16, scale=0.25
// ---------------------------------------------------------------------------
__global__ void mha_attn_kernel(const float* __restrict__ q, const float* __restrict__ k,
                                const float* __restrict__ v, float* __restrict__ out)
{
  long idx = (long)blockIdx.x * blockDim.x + threadIdx.x;
  if (idx >= (long)BATCH * NNODES * 8) return;
  const int h = idx % 8;
  const int n = (idx / 8) % NNODES;
  const int l = idx / (8 * NNODES);

  const long base = (long)n * ZDIM + h * 16;
  const float* qp = q + (long)l * NNODES * ZDIM + base;
  float qv[16];
#pragma unroll
  for (int d = 0; d < 16; ++d) qv[d] = qp[d];

  float sc[8], mx = -3.0e38f;
#pragma unroll
  for (int m = 0; m < 8; ++m) {
    const float* kp = k + (long)m * NNODES * ZDIM + base;
    float s = 0.f;
#pragma unroll
    for (int d = 0; d < 16; ++d) s += qv[d] * kp[d];
    s *= 0.25f;                // 1/sqrt(16)
    sc[m] = s;
    mx = fmaxf(mx, s);
  }
  float den = 0.f;
#pragma unroll
  for (int m = 0; m < 8; ++m) { sc[m] = __expf(sc[m] - mx); den += sc[m]; }
  const float inv = 1.0f / den;

  float o[16];
#pragma unroll
  for (int d = 0; d < 16; ++d) o[d] = 0.f;
#pragma unroll
  for (int m = 0; m < 8; ++m) {
    const float w = sc[m] * inv;
    const float* vp = v + (long)m * NNODES * ZDIM + base;
#pragma unroll
    for (int d = 0; d < 16; ++d) o[d] += w * vp[d];
  }
  float* op = out + (long)l * NNODES * ZDIM + base;
#pragma unroll
  for (int d = 0; d < 16; ++d) op[d] = o[d];
}

// ---------------------------------------------------------------------------
// [B,N,288] -> [B,288,N]
// ---------------------------------------------------------------------------
__global__ void transpose_kernel(const float* __restrict__ in, float* __restrict__ out)
{
  long idx = (long)blockIdx.x * blockDim.x + threadIdx.x;
  if (idx >= (long)BATCH * NNODES * CIN) return;
  const int c = idx % CIN;
  const int n = (idx / CIN) % NNODES;
  const int b = idx / ((long)CIN * NNODES);
  out[((long)b * CIN + c) * NNODES + n] = in[idx];
}

// ---------------------------------------------------------------------------
// Host-side launch helpers
// ---------------------------------------------------------------------------
static inline void gemm(hipStream_t st,
                        const float* A, long rsA, long csA, long bsA,
                        const float* B, long rsB, long csB, long bsB,
                        float* C, long ldc, long bsC,
                        const float* Cin, long ldcin, long bsCin,
                        float alpha, float beta, int M, int N, int K, int batch)
{
  dim3 grid((N + 63) / 64, (M + 127) / 128, batch);
  gemm_wmma_kernel<<<grid, 128, 0, st>>>(A, rsA, csA, bsA, B, rsB, csB, bsB,
                                         C, ldc, bsC, Cin, ldcin, bsCin,
                                         alpha, beta, M, N, K);
}

static inline void epilogue(hipStream_t st, float* C, const float* bias,
                            const float* res, long n, int cols, int act)
{
  epilogue_kernel<<<(unsigned)((n + 255) / 256), 256, 0, st>>>(C, bias, res, n, cols, act);
}

// d_gcn:  out[b,n,Z] = act( sum_j mats_j[b] @ ThetaJ(j) + bias ) (+ residual)
// mats: j=0:F ; j=1:Aq@F ; j=2:2Aq@x1-F ; j=3:Ah@F ; j=4:2Ah@x1-F
// ThetaJ(j) = rows (c*NM+j) of Theta -> base Theta + j*Z, row stride NM*Z.
static void d_gcn(hipStream_t st, const float* F, int c,
                  const float* Aq, const float* Ah,
                  const float* Theta, const float* bias, int Zout, int act,
                  const float* residual, float* out, float* T1, float* T2)
{
  const long bsF = (long)NNODES * c;
  const int  MF  = BATCH * NNODES;   // flattened rows for Theta GEMMs
  // j = 0
  gemm(st, F, c, 1, 0, Theta, (long)NM * Zout, 1, 0,
       out, Zout, 0, nullptr, 0, 0, 1.f, 0.f, MF, Zout, c, 1);
  int j = 1;
  const float* supports[2] = {Aq, Ah};
  for (int s = 0; s < 2; ++s) {
    const float* S = supports[s];
    // T1 = S @ F   (batched over B)
    gemm(st, S, NNODES, 1, 0, F, c, 1, bsF, T1, c, bsF,
         nullptr, 0, 0, 1.f, 0.f, NNODES, c, NNODES, BATCH);
    // out += T1 @ ThetaJ(j)
    gemm(st, T1, c, 1, 0, Theta + (long)j * Zout, (long)NM * Zout, 1, 0,
         out, Zout, 0, out, Zout, 0, 1.f, 1.f, MF, Zout, c, 1);
    ++j;
    // T2 = 2*S@T1 - F
    gemm(st, S, NNODES, 1, 0, T1, c, 1, bsF, T2, c, bsF,
         F, c, bsF, 2.f, -1.f, NNODES, c, NNODES, BATCH);
    // out += T2 @ ThetaJ(j)
    gemm(st, T2, c, 1, 0, Theta + (long)j * Zout, (long)NM * Zout, 1, 0,
         out, Zout, 0, out, Zout, 0, 1.f, 1.f, MF, Zout, c, 1);
    ++j;
  }
  epilogue(st, out, bias, residual, (long)MF * Zout, Zout, act);
}

// ---------------------------------------------------------------------------
// kernel_launch
// Input order assumed = JAX pytree flatten of setup_inputs() (sorted dict keys):
//  0:A_h 1:A_q 2:M 3:X 4:adj 5:conv_b 6:conv_w 7:dec.Theta 8:dec.bias
//  9:gat1.W 10:gat1.a_n 11:gat1.a_s 12:gat2.W 13:gat2.a_n 14:gat2.a_s
// 15:gnn1.Theta 16:gnn1.bias 17:gnn2.Theta 18:gnn2.bias
// 19..34: lstm[0..3] each {Whh,Wih,bhh,bih}
// 35:Wk 36:Wo 37:Wq 38:Wv 39:bk 40:bo 41:bq 42:bv
// ---------------------------------------------------------------------------
extern "C" void kernel_launch(void* const* d_in, const int* in_sizes, int n_in,
                              void* d_out, int out_size, void* d_ws, size_t ws_size,
                              hipStream_t stream)
{
  (void)in_sizes; (void)n_in; (void)out_size; (void)ws_size;
  const float* A_h   = (const float*)d_in[0];
  const float* A_q   = (const float*)d_in[1];
  const float* Mm    = (const float*)d_in[2];
  const float* X     = (const float*)d_in[3];
  const float* adj   = (const float*)d_in[4];
  const float* convb = (const float*)d_in[5];
  const float* convw = (const float*)d_in[6];
  const float* decT  = (const float*)d_in[7];
  const float* decB  = (const float*)d_in[8];
  const float* g1W   = (const float*)d_in[9];
  const float* g1an  = (const float*)d_in[10];
  const float* g1as  = (const float*)d_in[11];
  const float* g2W   = (const float*)d_in[12];
  const float* g2an  = (const float*)d_in[13];
  const float* g2as  = (const float*)d_in[14];
  const float* gnn1T = (const float*)d_in[15];
  const float* gnn1B = (const float*)d_in[16];
  const float* gnn2T = (const float*)d_in[17];
  const float* gnn2B = (const float*)d_in[18];
  const float* L[16];
  for (int i = 0; i < 16; ++i) L[i] = (const float*)d_in[19 + i];  // {Whh,Wih,bhh,bih} x4
  const float* Wk = (const float*)d_in[35];
  const float* Wo = (const float*)d_in[36];
  const float* Wq = (const float*)d_in[37];
  const float* Wv = (const float*)d_in[38];
  const float* bk = (const float*)d_in[39];
  const float* bo = (const float*)d_in[40];
  const float* bq = (const float*)d_in[41];
  const float* bv = (const float*)d_in[42];

  // ---- workspace layout (floats) ----
  float* w = (float*)d_ws;
  size_t off = 0;
  auto alloc = [&](size_t n) { float* p = w + off; off += n; return p; };
  float* xconv = alloc((size_t)BATCH * NNODES * 64);
  float* x1    = alloc((size_t)BATCH * NNODES * 288);
  float* Xs    = alloc((size_t)BATCH * NNODES * 288);
  float* Xs1   = alloc((size_t)BATCH * NNODES * ZDIM);
  float* Xem   = alloc((size_t)BATCH * NNODES * ZDIM);
  float* T1    = alloc((size_t)BATCH * NNODES * 288);
  float* T2    = alloc((size_t)BATCH * NNODES * 288);
  float* h1    = alloc((size_t)BATCH * NNODES * 256);
  float* h1o   = alloc((size_t)BATCH * NNODES * 256);
  float* h2    = alloc((size_t)BATCH * NNODES * ZDIM);
  float* Eraw  = alloc((size_t)BATCH * NNODES * ZDIM);
  float* ssv   = alloc((size_t)BATCH * NNODES);
  float* snv   = alloc((size_t)BATCH * NNODES);
  float* attb  = alloc((size_t)BATCH * NNODES * NNODES);   // union region below
  // q/k/v/attn_out/Xs2 reuse the attb region (disjoint lifetimes vs GAT)
  const size_t NEz = (size_t)BATCH * NNODES * ZDIM;
  float* qb  = attb;
  float* kb  = attb + NEz;
  float* vb  = attb + 2 * NEz;
  float* ao  = attb + 3 * NEz;
  float* Xs2 = attb + 4 * NEz;

  // ---- output layout ----
  float* out      = (float*)d_out;
  float* outXres  = out;                                         // [8,288,1024]
  float* outXatt  = out + (size_t)BATCH * CIN * NNODES;          // [8,1024,128]
  float* outEdge  = outXatt + NEz;                               // [8,1024,128]
  float* outApred = outEdge + NEz;                               // [1024,1024]

  const long NE = (long)NNODES * ZDIM;     // per-sample Edge/Z stride
  const int  MF = BATCH * NNODES;          // 8192 flattened rows

  // 1) conv + relu -> [B,N,64]
  conv_kernel<<<(unsigned)(((long)BATCH * NNODES * 64 + 255) / 256), 256, 0, stream>>>(
      X, convw, convb, xconv);

  // 2) BiLSTM layer 0 (I=64) -> x1[B,N,288], layer 1 (I=288) -> Xs[B,N,288]
  lstm_layer_kernel<<<2, 288, 0, stream>>>(xconv, 64,
      L[1], L[0], L[3], L[2],    // fwd: Wih,Whh,bih,bhh
      L[5], L[4], L[7], L[6],    // bwd
      x1);
  lstm_layer_kernel<<<2, 288, 0, stream>>>(x1, 288,
      L[9],  L[8],  L[11], L[10],
      L[13], L[12], L[15], L[14],
      Xs);

  // 3) gnn1: relu(d_gcn(Xs)) -> Xs1 [B,N,128]
  d_gcn(stream, Xs, 288, A_q, A_h, gnn1T, gnn1B, ZDIM, 1, nullptr, Xs1, T1, T2);
  // 4) gnn2: relu(d_gcn(Xs1)) + Xs1 -> Xem
  d_gcn(stream, Xs1, ZDIM, A_q, A_h, gnn2T, gnn2B, ZDIM, 1, Xs1, Xem, T1, T2);

  // 5) GAT (batched over 8 samples)
  // layer 1: h1 = Xs @ W1
  gemm(stream, Xs, 288, 1, (long)NNODES * 288, g1W, 256, 1, 0,
       h1, 256, (long)NNODES * 256, nullptr, 0, 0, 1.f, 0.f, NNODES, 256, 288, BATCH);
  sva_kernel<<<(MF + 255) / 256, 256, 0, stream>>>(h1, g1as, g1an, ssv, snv, 256);
  gat_att_kernel<<<dim3(NNODES, BATCH), 256, 0, stream>>>(ssv, snv, Mm, adj, attb);
  gemm(stream, attb, NNODES, 1, (long)NNODES * NNODES, h1, 256, 1, (long)NNODES * 256,
       h1o, 256, (long)NNODES * 256, nullptr, 0, 0, 1.f, 0.f, NNODES, 256, NNODES, BATCH);
  epilogue(stream, h1o, nullptr, nullptr, (long)MF * 256, 256, 3);  // elu
  // layer 2: h2 = h1o @ W2
  gemm(stream, h1o, 256, 1, (long)NNODES * 256, g2W, ZDIM, 1, 0,
       h2, ZDIM, NE, nullptr, 0, 0, 1.f, 0.f, NNODES, ZDIM, 256, BATCH);
  sva_kernel<<<(MF + 255) / 256, 256, 0, stream>>>(h2, g2as, g2an, ssv, snv, ZDIM);
  gat_att_kernel<<<dim3(NNODES, BATCH), 256, 0, stream>>>(ssv, snv, Mm, adj, attb);
  gemm(stream, attb, NNODES, 1, (long)NNODES * NNODES, h2, ZDIM, 1, NE,
       Eraw, ZDIM, NE, nullptr, 0, 0, 1.f, 0.f, NNODES, ZDIM, NNODES, BATCH);
  epilogue(stream, Eraw, nullptr, nullptr, (long)MF * ZDIM, ZDIM, 3);  // elu
  rownorm_kernel<<<MF, 128, 0, stream>>>(Eraw, outEdge, ZDIM);          // Edge_em

  // 6) A_pred = sigmoid(E0 @ E0^T)
  gemm(stream, outEdge, ZDIM, 1, 0, outEdge, 1, ZDIM, 0,
       outApred, NNODES, 0, nullptr, 0, 0, 1.f, 0.f, NNODES, NNODES, ZDIM, 1);
  epilogue(stream, outApred, nullptr, nullptr, (long)NNODES * NNODES, NNODES, 2);

  // 7) MHA: q,k from Xem; v from Edge_em   (W^T access: rsB=1, csB=Z)
  gemm(stream, Xem, ZDIM, 1, 0, Wq, 1, ZDIM, 0, qb, ZDIM, 0,
       nullptr, 0, 0, 1.f, 0.f, MF, ZDIM, ZDIM, 1);
  epilogue(stream, qb, bq, nullptr, (long)MF * ZDIM, ZDIM, 0);
  gemm(stream, Xem, ZDIM, 1, 0, Wk, 1, ZDIM, 0, kb, ZDIM, 0,
       nullptr, 0, 0, 1.f, 0.f, MF, ZDIM, ZDIM, 1);
  epilogue(stream, kb, bk, nullptr, (long)MF * ZDIM, ZDIM, 0);
  gemm(stream, outEdge, ZDIM, 1, 0, Wv, 1, ZDIM, 0, vb, ZDIM, 0,
       nullptr, 0, 0, 1.f, 0.f, MF, ZDIM, ZDIM, 1);
  epilogue(stream, vb, bv, nullptr, (long)MF * ZDIM, ZDIM, 0);
  mha_attn_kernel<<<((long)BATCH * NNODES * 8 + 255) / 256, 256, 0, stream>>>(qb, kb, vb, ao);
  gemm(stream, ao, ZDIM, 1, 0, Wo, 1, ZDIM, 0, outXatt, ZDIM, 0,
       nullptr, 0, 0, 1.f, 0.f, MF, ZDIM, ZDIM, 1);
  epilogue(stream, outXatt, bo, nullptr, (long)MF * ZDIM, ZDIM, 0);   // X_att

  // 8) decoder: X_s2 = d_gcn(X_att, linear) -> transpose to X_res
  d_gcn(stream, outXatt, ZDIM, A_q, A_h, decT, decB, CIN, 0, nullptr, Xs2, T1, T2);
  transpose_kernel<<<(unsigned)(((long)BATCH * NNODES * CIN + 255) / 256), 256, 0, stream>>>(
      Xs2, outXres);
}